// GAERGCN_13846974562749
// MI455X (gfx1250) — compile-verified
//
#include <hip/hip_runtime.h>

#define NN 50000      // N_NODES
#define NR 3          // N_REL
#define NE 500000     // N_EDGES
#define DIM 128       // IN_DIM == HID_DIM

typedef __attribute__((ext_vector_type(2))) float v2f;
typedef __attribute__((ext_vector_type(8))) float v8f;

// ---------------------------------------------------------------------------
// Zero a float buffer (used for degree counters).
__global__ __launch_bounds__(256) void rgcn_zero_f32(float* __restrict__ p, int n) {
    int i = blockIdx.x * 256 + threadIdx.x;
    if (i < n) p[i] = 0.0f;
}

// ---------------------------------------------------------------------------
// Count per-relation out/in degrees with f32 atomics (exact for counts < 2^24).
__global__ __launch_bounds__(256) void rgcn_count_deg(const int* __restrict__ src,
                                                      const int* __restrict__ dst,
                                                      float* __restrict__ outdeg,
                                                      float* __restrict__ indeg) {
    int i = blockIdx.x * 256 + threadIdx.x;      // over NR*NE flat edges
    if (i < NR * NE) {
        int r = i / NE;
        atomicAdd(outdeg + r * NN + src[i], 1.0f);
        atomicAdd(indeg  + r * NN + dst[i], 1.0f);
    }
}

// ---------------------------------------------------------------------------
// Initialize an accumulator [NN,DIM] with the relation-summed bias row.
__global__ __launch_bounds__(256) void rgcn_bias_init(float* __restrict__ acc,
                                                      const float* __restrict__ b) {
    int i = blockIdx.x * 256 + threadIdx.x;      // over NN*DIM
    if (i < NN * DIM) {
        int k = i & (DIM - 1);
        acc[i] = b[k] + b[DIM + k] + b[2 * DIM + k];
    }
}

// ---------------------------------------------------------------------------
// Z = (opt_relu(H) * rsqrt(max(outdeg,1)))[16-row tile] @ W   via f32 WMMA.
// Grid: NN/16 blocks of 256 threads (8 waves); wave w owns 16-col tile w.
// A tile staged in LDS (row pad 132 floats -> conflict-free ds reads).
template <bool RELU>
__global__ __launch_bounds__(256) void rgcn_gemm_wmma(const float* __restrict__ H,
                                                      const float* __restrict__ outdeg,
                                                      const float* __restrict__ W,
                                                      float* __restrict__ Z) {
    __shared__ float As[16][DIM + 4];
    const int row0 = blockIdx.x * 16;

    // Stage scaled (+ReLU) A tile: 2048 floats by 256 threads, coalesced rows.
    for (int t = threadIdx.x; t < 16 * DIM; t += 256) {
        const int r = t >> 7;
        const int col = t & (DIM - 1);
        const float rs = rsqrtf(fmaxf(outdeg[row0 + r], 1.0f));
        float v = H[(row0 + r) * DIM + col];
        if (RELU) v = fmaxf(v, 0.0f);
        As[r][col] = v * rs;
    }
    __syncthreads();

    const int wave = threadIdx.x >> 5;
    const int lane = threadIdx.x & 31;
    const int half = lane >> 4;      // 0: lanes 0-15, 1: lanes 16-31
    const int m    = lane & 15;      // M index (A) / N index (B,C)
    const int col0 = wave * 16;

    v8f acc = {};
    #pragma unroll
    for (int k0 = 0; k0 < DIM; k0 += 4) {
        const int ka = k0 + 2 * half;
        v2f a, b;
        // A 16x4 f32 layout: VGPR0 -> K=ka, VGPR1 -> K=ka+1 (halves offset K by 2)
        a.x = As[m][ka];
        a.y = As[m][ka + 1];
        // B 4x16 f32 layout: row K striped across lanes within a VGPR
        b.x = W[ka * DIM + col0 + m];
        b.y = W[(ka + 1) * DIM + col0 + m];
        acc = __builtin_amdgcn_wmma_f32_16x16x4_f32(
            /*neg_a=*/false, a, /*neg_b=*/false, b,
            /*c_mod=*/(short)0, acc, /*reuse_a=*/false, /*reuse_b=*/false);
    }

    // C/D layout: VGPR v -> M = v + 8*half, N = lane&15
    #pragma unroll
    for (int v = 0; v < 8; ++v) {
        Z[(row0 + v + 8 * half) * DIM + col0 + m] = acc[v];
    }
}

// ---------------------------------------------------------------------------
// acc[dst] += Z[src] * rsqrt(max(indeg[dst],1)).  One wave per edge,
// each lane handles a contiguous float4 chunk (coalesced 512B gather,
// 4 contiguous f32 atomics per lane).
__global__ __launch_bounds__(256) void rgcn_scatter(const float* __restrict__ Z,
                                                    const int* __restrict__ src,
                                                    const int* __restrict__ dst,
                                                    const float* __restrict__ indeg,
                                                    float* __restrict__ acc) {
    const long long idx = (long long)blockIdx.x * 256 + threadIdx.x;
    const int e = (int)(idx >> 5);
    const int lane = (int)(idx & 31);
    if (e >= NE) return;
    const int s = src[e];
    const int d = dst[e];
    const float sc = rsqrtf(fmaxf(indeg[d], 1.0f));
    const float4 v = *(const float4*)(Z + (long long)s * DIM + lane * 4);
    float* o = acc + (long long)d * DIM + lane * 4;
    atomicAdd(o + 0, v.x * sc);
    atomicAdd(o + 1, v.y * sc);
    atomicAdd(o + 2, v.z * sc);
    atomicAdd(o + 3, v.w * sc);
}

// ---------------------------------------------------------------------------
extern "C" void kernel_launch(void* const* d_in, const int* in_sizes, int n_in,
                              void* d_out, int out_size, void* d_ws, size_t ws_size,
                              hipStream_t stream) {
    const float* x    = (const float*)d_in[0];
    const int*   esrc = (const int*)  d_in[1];   // [NR, NE]
    const int*   edst = (const int*)  d_in[2];   // [NR, NE]
    const float* W1   = (const float*)d_in[3];   // [NR, DIM, DIM]
    const float* b1   = (const float*)d_in[4];   // [NR, DIM]
    const float* W2   = (const float*)d_in[5];
    const float* b2   = (const float*)d_in[6];
    float*       out  = (float*)d_out;           // [NN, DIM]

    float* ws = (float*)d_ws;
    float* Z      = ws;                                   // NN*DIM floats
    float* H1     = Z + (size_t)NN * DIM;                 // NN*DIM floats
    float* outdeg = H1 + (size_t)NN * DIM;                // NR*NN floats
    float* indeg  = outdeg + (size_t)NR * NN;             // NR*NN floats

    const int degN    = 2 * NR * NN;
    const int gemmGrd = NN / 16;                          // 3125 (exact)
    const int scatGrd = (NE * 32) / 256;                  // 62500 (exact)

    // Degrees (shared by both layers).
    rgcn_zero_f32<<<(degN + 255) / 256, 256, 0, stream>>>(outdeg, degN);
    rgcn_count_deg<<<(NR * NE + 255) / 256, 256, 0, stream>>>(esrc, edst, outdeg, indeg);

    // -------- Layer 1: H1 = sum_r GraphConv(x; W1_r, b1_r) --------
    rgcn_bias_init<<<(NN * DIM + 255) / 256, 256, 0, stream>>>(H1, b1);
    for (int r = 0; r < NR; ++r) {
        rgcn_gemm_wmma<false><<<gemmGrd, 256, 0, stream>>>(
            x, outdeg + r * NN, W1 + (size_t)r * DIM * DIM, Z);
        rgcn_scatter<<<scatGrd, 256, 0, stream>>>(
            Z, esrc + r * NE, edst + r * NE, indeg + r * NN, H1);
    }

    // -------- Layer 2: out = sum_r GraphConv(relu(H1); W2_r, b2_r) --------
    // ReLU is folded into the layer-2 GEMM's A staging load.
    rgcn_bias_init<<<(NN * DIM + 255) / 256, 256, 0, stream>>>(out, b2);
    for (int r = 0; r < NR; ++r) {
        rgcn_gemm_wmma<true><<<gemmGrd, 256, 0, stream>>>(
            H1, outdeg + r * NN, W2 + (size_t)r * DIM * DIM, Z);
        rgcn_scatter<<<scatGrd, 256, 0, stream>>>(
            Z, esrc + r * NE, edst + r * NE, indeg + r * NN, out);
    }
}